// P2iBlock_53764400611481
// MI455X (gfx1250) — compile-verified
//
#include <hip/hip_runtime.h>
#include <hip/hip_bf16.h>
#include <math.h>

// Problem constants (match reference)
#define KNN 32
#define HH  36
#define WW  36
#define BB  8
#define NN  8192
#define SS  2048

#define WAVES 4            // waves per block
#define QW    16           // queries per wave (WMMA M dim)
#define QB    (WAVES*QW)   // 64 queries per block
#define PTILE 64           // points per main-loop iteration (4 WMMAs)

typedef __attribute__((ext_vector_type(2))) float v2f;
typedef __attribute__((ext_vector_type(8))) float v8f;

__global__ __launch_bounds__(WAVES * 32)
void knn_sphere_wmma_kernel(const float* __restrict__ xyz,      // [B,N,3]
                            const float* __restrict__ new_xyz,  // [B,S,3]
                            float* __restrict__ hist,           // [B,S,H,W]
                            float* __restrict__ gfeat,          // [B,S,K,6]
                            float* __restrict__ rho_out)        // [B,S,K]
{
    __shared__ float s_qq[WAVES][QW];            // query squared norms
    __shared__ float s_dt[WAVES][QW][PTILE];     // 16x64 distance tile per wave
    __shared__ float s_cd[WAVES][QW][KNN];       // candidate distances
    __shared__ int   s_ci[WAVES][QW][KNN];       // candidate indices
    __shared__ float s_hist[WAVES][HH * WW];     // per-query histogram staging

    const int tid  = threadIdx.x;
    const int wv   = tid >> 5;
    const int lane = tid & 31;
    const int half = lane >> 4;   // 0: carries (x,y) rows K0,K1 ; 1: carries (z,pad) rows K2,K3
    const int l16  = lane & 15;

    const int sPerBlk = SS / QB;                          // 32
    const int b  = blockIdx.x / sPerBlk;
    const int s0 = (blockIdx.x % sPerBlk) * QB + wv * QW; // first query of this wave

    // ---- WMMA A operand: 16 queries x 4 (K3 zero-padded), 32-bit A layout 16x4 ----
    const float* qp = new_xyz + ((size_t)b * SS + s0 + l16) * 3;
    const float qx = qp[0], qy = qp[1], qz = qp[2];
    v2f a;
    a.x = half ? qz  : qx;
    a.y = half ? 0.f : qy;
    if (lane < QW) s_qq[wv][lane] = qx * qx + qy * qy + qz * qz;
    __syncthreads();

    float qqr[8];
#pragma unroll
    for (int r = 0; r < 8; ++r) qqr[r] = s_qq[wv][r + 8 * half];

    // init candidate lists to +inf
    for (int k = lane; k < QW * KNN; k += 32) {
        (&s_cd[wv][0][0])[k] = 3.4e38f;
        (&s_ci[wv][0][0])[k] = 0;
    }
    __syncthreads();

    float wmax = 3.4e38f;  // worst (largest) distance currently kept (lane owns query=lane)
    int   wpos = 0;

    const float* xb = xyz + (size_t)b * NN * 3;

    for (int n0 = 0; n0 < NN; n0 += PTILE) {
        // prefetch the whole next 64-point chunk (768B, lane-strided across cachelines)
        if (n0 + PTILE < NN)
            __builtin_prefetch(xb + (size_t)(n0 + PTILE + lane * 2) * 3, 0, 1);

        // ---- 4 x WMMA: 16 queries x 64 points of dot products ----
        v8f   d[4];
        float pn[4];
#pragma unroll
        for (int t = 0; t < 4; ++t) {
            const float* pp = xb + (size_t)(n0 + t * 16 + l16) * 3;
            const float px = pp[0], py = pp[1], pz = pp[2];
            v2f bop;                       // 4x16 B operand, striped like A
            bop.x = half ? pz  : px;
            bop.y = half ? 0.f : py;
            pn[t] = px * px + py * py + pz * pz;
            v8f c = {};
            d[t] = __builtin_amdgcn_wmma_f32_16x16x4_f32(false, a, false, bop,
                                                         (short)0, c, false, false);
        }

        // dist = |q|^2 + |p|^2 - 2 q.p ; stage tile to LDS (M = r + 8*half, N = t*16 + l16)
#pragma unroll
        for (int t = 0; t < 4; ++t) {
#pragma unroll
            for (int r = 0; r < 8; ++r)
                s_dt[wv][r + 8 * half][t * 16 + l16] = qqr[r] + pn[t] - 2.0f * d[t][r];
        }
        __syncthreads();

        // ---- per-query top-K insert: lane m exclusively owns query m ----
        if (lane < QW) {
            for (int j = 0; j < PTILE; ++j) {
                const float dd = s_dt[wv][lane][j];
                if (dd < wmax) {
                    s_cd[wv][lane][wpos] = dd;
                    s_ci[wv][lane][wpos] = n0 + j;
                    float m = -3.4e38f; int mp = 0;
#pragma unroll 4
                    for (int k = 0; k < KNN; ++k) {
                        const float v = s_cd[wv][lane][k];
                        if (v > m) { m = v; mp = k; }
                    }
                    wmax = m; wpos = mp;
                }
            }
        }
        __syncthreads();
    }

    // ---- sort candidates ascending (reference top_k order) ----
    if (lane < QW) {
        for (int i = 0; i < KNN - 1; ++i) {
            int mp = i; float mv = s_cd[wv][lane][i];
            for (int k = i + 1; k < KNN; ++k) {
                const float v = s_cd[wv][lane][k];
                if (v < mv) { mv = v; mp = k; }
            }
            const float td = s_cd[wv][lane][i];
            s_cd[wv][lane][i] = s_cd[wv][lane][mp]; s_cd[wv][lane][mp] = td;
            const int ti = s_ci[wv][lane][i];
            s_ci[wv][lane][i] = s_ci[wv][lane][mp]; s_ci[wv][lane][mp] = ti;
        }
    }
    __syncthreads();

    // ---- spherical features + histogram: one query per iteration, lane = neighbor ----
    const float PI = 3.14159265358979323846f;
    for (int q = 0; q < QW; ++q) {
        const int s = s0 + q;
        const float* qc = new_xyz + ((size_t)b * SS + s) * 3;
        const float cqx = qc[0], cqy = qc[1], cqz = qc[2];

        const int pidx = s_ci[wv][q][lane];
        const float* pr = xb + (size_t)pidx * 3;
        const float rx = pr[0] - cqx, ry = pr[1] - cqy, rz = pr[2] - cqz;
        const float rho = sqrtf(rx * rx + ry * ry + rz * rz);
        const bool safe = rho > 0.f;
        float ratio = safe ? rz / rho : 0.f;
        ratio = fminf(1.f, fmaxf(-1.f, ratio));
        const float theta = safe ? acosf(ratio) / PI : 0.f;
        const float phi   = atan2f(ry, rx) / (2.f * PI) + 0.5f;

        float rmax = rho;                   // wave32 max-reduce over the 32 neighbors
#pragma unroll
        for (int off = 16; off > 0; off >>= 1)
            rmax = fmaxf(rmax, __shfl_xor(rmax, off, 32));
        const float weight = fmaxf(0.5f - rho / (2.f * rmax), 0.f) + 0.5f;

        int u = (int)floorf(theta * (HH - 1)); u = min(HH - 1, max(0, u));
        int v = (int)floorf(phi   * (WW - 1)); v = min(WW - 1, max(0, v));

        for (int j = lane; j < HH * WW; j += 32) s_hist[wv][j] = 0.f;
        __syncthreads();
        atomicAdd(&s_hist[wv][u * WW + v], weight);   // ds_add_f32
        __syncthreads();

        float* ho = hist + ((size_t)b * SS + s) * (HH * WW);
        for (int j = lane; j < HH * WW; j += 32)
            __builtin_nontemporal_store(s_hist[wv][j], ho + j);

        float* go = gfeat + (((size_t)b * SS + s) * KNN + lane) * 6;
        __builtin_nontemporal_store(rho,   go + 0);
        __builtin_nontemporal_store(theta, go + 1);
        __builtin_nontemporal_store(phi,   go + 2);
        __builtin_nontemporal_store(rx,    go + 3);
        __builtin_nontemporal_store(ry,    go + 4);
        __builtin_nontemporal_store(rz,    go + 5);
        __builtin_nontemporal_store(rho, rho_out + ((size_t)b * SS + s) * KNN + lane);
        __syncthreads();
    }
}

extern "C" void kernel_launch(void* const* d_in, const int* in_sizes, int n_in,
                              void* d_out, int out_size, void* d_ws, size_t ws_size,
                              hipStream_t stream) {
    const float* xyz     = (const float*)d_in[0];   // [B,N,3]
    const float* new_xyz = (const float*)d_in[1];   // [B,S,3]

    float* out   = (float*)d_out;
    float* hist  = out;                                       // B*S*H*W
    float* gfeat = hist + (size_t)BB * SS * HH * WW;          // B*S*K*6
    float* rho   = gfeat + (size_t)BB * SS * KNN * 6;         // B*S*K

    dim3 grid(BB * (SS / QB));     // 8 * 32 = 256 blocks
    dim3 block(WAVES * 32);        // 128 threads = 4 wave32
    hipLaunchKernelGGL(knn_sphere_wmma_kernel, grid, block, 0, stream,
                       xyz, new_xyz, hist, gfeat, rho);
}